// MultiHeadAttention_24215025615631
// MI455X (gfx1250) — compile-verified
//
#include <hip/hip_runtime.h>

typedef float v2f __attribute__((ext_vector_type(2)));
typedef float v8f __attribute__((ext_vector_type(8)));

// D = A(16x4 f32) * B(4x16 f32) + C, full fp32 precision (matches reference).
#define WMMA_F32(a, b, c) \
  __builtin_amdgcn_wmma_f32_16x16x4_f32(false, (a), false, (b), (short)0, (c), false, false)

// ---------------------------------------------------------------------------
// Y[m,n] = sum_k X[m,k] * W[n,k]   (y = x @ W.T),  X: 8192x512, W: 512x512.
// 8 waves/block, each wave owns a 16x64 output tile (4 f32 accumulators).
// A and B both load as contiguous float2 (K contiguous for both since B=W^T).
// ---------------------------------------------------------------------------
__global__ __launch_bounds__(256) void gemm_xwt_512(const float* __restrict__ X,
                                                    const float* __restrict__ W,
                                                    float* __restrict__ Y) {
  const int lane = threadIdx.x;
  const int lm = lane & 15, kh = lane >> 4;
  const int mt = blockIdx.y * 8 + threadIdx.y;   // m-tile 0..511
  const int n0 = blockIdx.x * 64;
  const float* xrow = X + (size_t)(mt * 16 + lm) * 512 + 2 * kh;
  const float* wrow = W + (size_t)(n0 + lm) * 512 + 2 * kh;
  v8f acc[4] = {};
#pragma unroll 4
  for (int k0 = 0; k0 < 512; k0 += 4) {
    v2f a = *(const v2f*)(xrow + k0);
#pragma unroll
    for (int t = 0; t < 4; ++t) {
      v2f b = *(const v2f*)(wrow + (size_t)t * 16 * 512 + k0);
      acc[t] = WMMA_F32(a, b, acc[t]);
    }
  }
#pragma unroll
  for (int t = 0; t < 4; ++t)
#pragma unroll
    for (int r = 0; r < 8; ++r)
      Y[(size_t)(mt * 16 + r + 8 * kh) * 512 + n0 + t * 16 + lm] = acc[t][r];
}

// ---------------------------------------------------------------------------
// Fused S = scale * Qh @ Kh^T  ->  softmax rows  ->  P (written once to HBM).
// Block = 256 thr (8 waves) handles one (b, head, 16-row) stripe: each wave
// computes a 16x128 S chunk with WMMA, dumps to 64KB LDS, then the block does
// a 3-pass row softmax and streams normalized P to d_out with float4 stores.
// Qh/Kh are contiguous 1024x64 blocks thanks to the reference's .view reshape.
// ---------------------------------------------------------------------------
__global__ __launch_bounds__(256) void attn_score_softmax(const float* __restrict__ qp,
                                                          const float* __restrict__ kp,
                                                          float* __restrict__ P) {
  __shared__ float sS[16 * 1024];  // 64 KB of the 320 KB WGP LDS
  const int lane = threadIdx.x;
  const int w = threadIdx.y;  // wave 0..7 -> columns [128w, 128w+128)
  const int lm = lane & 15, kh = lane >> 4;
  const int i0 = blockIdx.x * 16;
  const int n = blockIdx.y, b = blockIdx.z;
  const float* Qh = qp + (size_t)b * 524288 + (size_t)n * 65536;
  const float* Kh = kp + (size_t)b * 524288 + (size_t)n * 65536;
  const float* qrow = Qh + (size_t)(i0 + lm) * 64 + 2 * kh;
  const float* krow = Kh + (size_t)(w * 128 + lm) * 64 + 2 * kh;
  v8f acc[8] = {};
#pragma unroll
  for (int k0 = 0; k0 < 64; k0 += 4) {
    v2f a = *(const v2f*)(qrow + k0);
#pragma unroll
    for (int t = 0; t < 8; ++t) {
      v2f bm = *(const v2f*)(krow + t * 1024 + k0);  // (128w+16t+lm)*64
      acc[t] = WMMA_F32(a, bm, acc[t]);
    }
  }
  const float scale = 0.04419417382415922f;  // 512^-0.5 (width, per reference)
#pragma unroll
  for (int t = 0; t < 8; ++t)
#pragma unroll
    for (int r = 0; r < 8; ++r)
      sS[(r + 8 * kh) * 1024 + w * 128 + t * 16 + lm] = acc[t][r] * scale;
  __syncthreads();

  // Row softmax: 16 threads per row, 64 elements each; reduce via shfl_xor(16).
  const int tid = w * 32 + lane;
  const int row = tid >> 4, sub = tid & 15;
  float* srow = sS + row * 1024 + sub * 64;
  float mx = -3.402823466e38f;
#pragma unroll
  for (int c = 0; c < 64; ++c) mx = fmaxf(mx, srow[c]);
#pragma unroll
  for (int off = 1; off < 16; off <<= 1) mx = fmaxf(mx, __shfl_xor(mx, off, 16));
  float sum = 0.f;
#pragma unroll
  for (int c = 0; c < 64; ++c) {
    float e = __expf(srow[c] - mx);
    srow[c] = e;
    sum += e;
  }
#pragma unroll
  for (int off = 1; off < 16; off <<= 1) sum += __shfl_xor(sum, off, 16);
  const float inv = 1.0f / sum;
  float* prow = P + ((size_t)(b * 8 + n) * 1024 + i0 + row) * 1024 + sub * 64;
#pragma unroll
  for (int c4 = 0; c4 < 16; ++c4) {
    float4 v = *(const float4*)(srow + c4 * 4);
    v.x *= inv; v.y *= inv; v.z *= inv; v.w *= inv;
    *(float4*)(prow + c4 * 4) = v;
  }
}

// ---------------------------------------------------------------------------
// X = P(1024x1024) @ Vh(1024x64) per (b, head). Each wave: full 16x64 tile.
// Output written directly in (b, i, n*64+d) order so Wo projection is a GEMM.
// ---------------------------------------------------------------------------
__global__ __launch_bounds__(128) void attn_pv(const float* __restrict__ P,
                                               const float* __restrict__ vp,
                                               float* __restrict__ Xp) {
  const int lane = threadIdx.x;
  const int lm = lane & 15, kh = lane >> 4;
  const int mt = blockIdx.x * 4 + threadIdx.y;  // row-tile 0..63
  const int n = blockIdx.y, b = blockIdx.z;
  const float* Ph = P + (size_t)(b * 8 + n) * 1024 * 1024;
  const float* Vh = vp + (size_t)b * 524288 + (size_t)n * 65536;
  const float* prow = Ph + (size_t)(mt * 16 + lm) * 1024 + 2 * kh;
  v8f acc[4] = {};
#pragma unroll 2
  for (int k0 = 0; k0 < 1024; k0 += 4) {
    v2f a = *(const v2f*)(prow + k0);
    const int kb = (k0 + 2 * kh) * 64 + lm;  // B[k][d] = Vh[k*64 + d], no transpose
#pragma unroll
    for (int t = 0; t < 4; ++t) {
      v2f bm;
      bm.x = Vh[kb + t * 16];
      bm.y = Vh[kb + 64 + t * 16];
      acc[t] = WMMA_F32(a, bm, acc[t]);
    }
  }
  float* xb = Xp + (size_t)b * 524288 + (size_t)n * 64;
#pragma unroll
  for (int t = 0; t < 4; ++t)
#pragma unroll
    for (int r = 0; r < 8; ++r)
      xb[(size_t)(mt * 16 + r + 8 * kh) * 512 + t * 16 + lm] = acc[t][r];
}

extern "C" void kernel_launch(void* const* d_in, const int* in_sizes, int n_in,
                              void* d_out, int out_size, void* d_ws, size_t ws_size,
                              hipStream_t stream) {
  (void)in_sizes; (void)n_in; (void)out_size; (void)ws_size;
  const float* q  = (const float*)d_in[0];
  const float* k  = (const float*)d_in[1];
  const float* v  = (const float*)d_in[2];
  const float* Wq = (const float*)d_in[3];
  const float* Wk = (const float*)d_in[4];
  const float* Wv = (const float*)d_in[5];
  const float* Wo = (const float*)d_in[6];

  float* x_out = (float*)d_out;                     // (8,1024,512)
  float* P     = x_out + (size_t)8192 * 512;        // (8,8,1024,1024) attn_weight

  float* qp = (float*)d_ws;                         // 16 MB each
  float* kp = qp + (size_t)8192 * 512;
  float* vp = kp + (size_t)8192 * 512;
  float* xp = vp + (size_t)8192 * 512;              // total 64 MB scratch

  const dim3 gblk(32, 8), ggrd(8, 64);
  gemm_xwt_512<<<ggrd, gblk, 0, stream>>>(q, Wq, qp);
  gemm_xwt_512<<<ggrd, gblk, 0, stream>>>(k, Wk, kp);
  gemm_xwt_512<<<ggrd, gblk, 0, stream>>>(v, Wv, vp);

  attn_score_softmax<<<dim3(64, 8, 8), dim3(32, 8), 0, stream>>>(qp, kp, P);
  attn_pv<<<dim3(16, 8, 8), dim3(32, 4), 0, stream>>>(P, vp, xp);
  gemm_xwt_512<<<ggrd, gblk, 0, stream>>>(xp, Wo, x_out);
}